// NeuralFractionalODE_1228360647372
// MI455X (gfx1250) — compile-verified
//
#include <hip/hip_runtime.h>
#include <math.h>

// ---------------------------------------------------------------------------
// NeuralFractionalODE for MI455X (gfx1250).
// sol[b,s,:] = y0[b,:] + dt * P[s,:]  with P = exclusive cumsum of the
// batch-invariant MLP table rhs[s,:].  conv+mean collapses analytically to
// column sums of sol plus <=5-element edge corrections.  Dominant cost is the
// 134 MB store of sol (~5.8us at 23.3 TB/s); MLP done with fp32 WMMA.
// ---------------------------------------------------------------------------

typedef float v2f __attribute__((ext_vector_type(2)));
typedef float v8f __attribute__((ext_vector_type(8)));

#define S_LEN 8192
#define D_DIM 64

// ===========================================================================
// Kernel 1: rhs[s,:] = MLP(t_{s+1}) for s = 0..8191 (row 8191 is junk/unused).
// One wave per 16-row tile, 2 waves (64 threads) per block, fp32 WMMA 16x16x4.
// ===========================================================================
__global__ __launch_bounds__(64) void mlp_rhs_kernel(
    const float* __restrict__ t,
    const float* __restrict__ W1, const float* __restrict__ b1,
    const float* __restrict__ W2, const float* __restrict__ b2,
    const float* __restrict__ W3, const float* __restrict__ b3,
    const float* __restrict__ W4, const float* __restrict__ b4,
    float* __restrict__ RHS)
{
    __shared__ float sW2[64 * 128];
    __shared__ float sW3[128 * 64];
    __shared__ float sW4[64 * 64];
    __shared__ float sB2[128];
    __shared__ float sB3[64];
    __shared__ float sB4[64];
    __shared__ float sH1[2][16 * 64];
    __shared__ float sH2[2][16 * 128];
    __shared__ float sH3[2][16 * 64];

    const int tid = threadIdx.x;
    for (int i = tid; i < 64 * 128; i += 64) sW2[i] = W2[i];
    for (int i = tid; i < 128 * 64; i += 64) sW3[i] = W3[i];
    for (int i = tid; i < 64 * 64;  i += 64) sW4[i] = W4[i];
    sB2[tid] = b2[tid];
    sB2[tid + 64] = b2[tid + 64];
    sB3[tid] = b3[tid];
    sB4[tid] = b4[tid];
    __syncthreads();

    const int wave = tid >> 5;
    const int lane = tid & 31;
    const int c = lane & 15;   // column within 16-wide fragment
    const int g = lane >> 4;   // lane half (selects K/M pairs per ISA layout)

    const int tile = blockIdx.x * 2 + wave;   // 0..511
    const int row0 = tile * 16;

    float* h1 = sH1[wave];
    float* h2 = sH2[wave];
    float* h3 = sH3[wave];

    // Layer 1: input is [t, 0...0] -> only W1 row 0 contributes.
    for (int i = lane; i < 16 * 64; i += 32) {
        const int m = i >> 6, j = i & 63;
        const float tv = t[row0 + m + 1];           // in-bounds of [B,S] buffer
        h1[i] = tanhf(tv * W1[j] + b1[j]);
    }
    __syncthreads();

    // Layer 2: h2[16x128] = tanh(h1[16x64] @ W2[64x128] + b2)
    for (int n = 0; n < 8; ++n) {
        v8f acc;
        const float bias = sB2[n * 16 + c];
        #pragma unroll
        for (int v = 0; v < 8; ++v) acc[v] = bias;
        #pragma unroll
        for (int kk = 0; kk < 16; ++kk) {
            v2f a, b;
            a[0] = h1[c * 64 + 4 * kk + 2 * g + 0];
            a[1] = h1[c * 64 + 4 * kk + 2 * g + 1];
            b[0] = sW2[(4 * kk + 2 * g + 0) * 128 + n * 16 + c];
            b[1] = sW2[(4 * kk + 2 * g + 1) * 128 + n * 16 + c];
            acc = __builtin_amdgcn_wmma_f32_16x16x4_f32(
                false, a, false, b, (short)0, acc, false, false);
        }
        #pragma unroll
        for (int v = 0; v < 8; ++v)
            h2[(v + 8 * g) * 128 + n * 16 + c] = tanhf(acc[v]);
    }
    __syncthreads();

    // Layer 3: h3[16x64] = tanh(h2[16x128] @ W3[128x64] + b3)
    for (int n = 0; n < 4; ++n) {
        v8f acc;
        const float bias = sB3[n * 16 + c];
        #pragma unroll
        for (int v = 0; v < 8; ++v) acc[v] = bias;
        #pragma unroll
        for (int kk = 0; kk < 32; ++kk) {
            v2f a, b;
            a[0] = h2[c * 128 + 4 * kk + 2 * g + 0];
            a[1] = h2[c * 128 + 4 * kk + 2 * g + 1];
            b[0] = sW3[(4 * kk + 2 * g + 0) * 64 + n * 16 + c];
            b[1] = sW3[(4 * kk + 2 * g + 1) * 64 + n * 16 + c];
            acc = __builtin_amdgcn_wmma_f32_16x16x4_f32(
                false, a, false, b, (short)0, acc, false, false);
        }
        #pragma unroll
        for (int v = 0; v < 8; ++v)
            h3[(v + 8 * g) * 64 + n * 16 + c] = tanhf(acc[v]);
    }
    __syncthreads();

    // Layer 4: RHS[16x64] = h3[16x64] @ W4[64x64] + b4  (no activation)
    for (int n = 0; n < 4; ++n) {
        v8f acc;
        const float bias = sB4[n * 16 + c];
        #pragma unroll
        for (int v = 0; v < 8; ++v) acc[v] = bias;
        #pragma unroll
        for (int kk = 0; kk < 16; ++kk) {
            v2f a, b;
            a[0] = h3[c * 64 + 4 * kk + 2 * g + 0];
            a[1] = h3[c * 64 + 4 * kk + 2 * g + 1];
            b[0] = sW4[(4 * kk + 2 * g + 0) * 64 + n * 16 + c];
            b[1] = sW4[(4 * kk + 2 * g + 1) * 64 + n * 16 + c];
            acc = __builtin_amdgcn_wmma_f32_16x16x4_f32(
                false, a, false, b, (short)0, acc, false, false);
        }
        #pragma unroll
        for (int v = 0; v < 8; ++v)
            RHS[(row0 + v + 8 * g) * 64 + n * 16 + c] = acc[v];
    }
}

// ===========================================================================
// Kernel 2: per-chunk (128 rows) exclusive scan of RHS along s, per channel.
// Also emits chunk totals and sum-of-local-prefixes (for the analytic mean).
// ===========================================================================
__global__ __launch_bounds__(64) void scan_local_kernel(
    const float* __restrict__ RHS, float* __restrict__ Pbuf,
    float* __restrict__ CSum, float* __restrict__ CExcl)
{
    const int d = threadIdx.x;
    const int chunk = blockIdx.x;
    const int base = chunk * 128;
    float run = 0.0f, excl = 0.0f;
    for (int r = 0; r < 128; ++r) {
        const int s = base + r;
        Pbuf[s * 64 + d] = run;     // exclusive local prefix
        excl += run;
        const float v = (s < S_LEN - 1) ? RHS[s * 64 + d] : 0.0f; // row 8191 junk
        run += v;
    }
    CSum[chunk * 64 + d] = run;
    CExcl[chunk * 64 + d] = excl;
}

// ===========================================================================
// Kernel 3: exclusive scan of chunk totals; SC[d] = sum_s P[s,d].
// ===========================================================================
__global__ __launch_bounds__(64) void scan_chunks_kernel(
    const float* __restrict__ CSum, const float* __restrict__ CExcl,
    float* __restrict__ Off, float* __restrict__ SC)
{
    const int d = threadIdx.x;
    float run = 0.0f, sc = 0.0f;
    for (int c = 0; c < 64; ++c) {
        Off[c * 64 + d] = run;
        sc += 128.0f * run + CExcl[c * 64 + d];
        run += CSum[c * 64 + d];
    }
    SC[d] = sc;
}

// ===========================================================================
// Kernel 4: streaming writer (134 MB): sol[b,s,:] = y0[b,:] + dt*(Off+Plocal).
// One float4 store per thread -> global_store_b128 at HBM rate.
// ===========================================================================
__global__ __launch_bounds__(256) void write_sol_kernel(
    const float* __restrict__ t, const float* __restrict__ y0,
    const float* __restrict__ Pbuf, const float* __restrict__ Off,
    float* __restrict__ out)
{
    const float dt = t[1] - t[0];
    const unsigned idx = blockIdx.x * 256u + threadIdx.x;   // float4 index
    const int d4 = idx & 15;
    const int s  = (idx >> 4) & (S_LEN - 1);
    const int b  = idx >> 17;

    const float4 p = ((const float4*)Pbuf)[s * 16 + d4];
    const float4 o = ((const float4*)Off)[(s >> 7) * 16 + d4];
    const float4 y = ((const float4*)y0)[b * 16 + d4];

    float4 r;
    r.x = y.x + dt * (p.x + o.x);
    r.y = y.y + dt * (p.y + o.y);
    r.z = y.z + dt * (p.z + o.z);
    r.w = y.w + dt * (p.w + o.w);
    ((float4*)out)[idx] = r;
}

// ===========================================================================
// Kernel 5: analytic conv-mean features + hurst head (one block).
// mean_s conv_k(x)[b,o] = (1/S) * sum_i sum_taps K[o,i,k]*(T[b,i]-edge(d))
// with T[b,i] = S*y0 + dt*SC[i] and edge = head/tail partial sums.
// ===========================================================================
__global__ __launch_bounds__(256) void features_kernel(
    const float* __restrict__ t, const float* __restrict__ y0,
    const float* __restrict__ Pbuf, const float* __restrict__ Off,
    const float* __restrict__ SC,
    const float* __restrict__ K3,  const float* __restrict__ c3,
    const float* __restrict__ K5,  const float* __restrict__ c5,
    const float* __restrict__ K7,  const float* __restrict__ c7,
    const float* __restrict__ K11, const float* __restrict__ c11,
    const float* __restrict__ H1,  const float* __restrict__ hb1,
    const float* __restrict__ H2,  const float* __restrict__ hb2,
    float* __restrict__ hurst)
{
    __shared__ float sSC[64];
    __shared__ float sHP[5 * 64];   // HP[u][i] = sum_{v<=u} Ptot[v][i]
    __shared__ float sTP[5 * 64];   // TP[u][i] = sum_{v<=u} Ptot[S-1-v][i]
    __shared__ float sFeat[64 * 64];

    const int tid = threadIdx.x;
    const float dt = t[1] - t[0];

    if (tid < 64) {
        const int d = tid;
        sSC[d] = SC[d];
        float hp = 0.0f;
        #pragma unroll
        for (int u = 0; u < 5; ++u) {            // chunk 0 => Off == 0
            hp += Pbuf[u * 64 + d];
            sHP[u * 64 + d] = hp;
        }
        const float off63 = Off[63 * 64 + d];
        float tp = 0.0f;
        #pragma unroll
        for (int u = 0; u < 5; ++u) {
            tp += off63 + Pbuf[(S_LEN - 1 - u) * 64 + d];
            sTP[u * 64 + d] = tp;
        }
    }
    __syncthreads();

    for (int pair = tid; pair < 64 * 64; pair += 256) {
        const int b = pair >> 6;
        const int o = pair & 63;
        float acc = 0.0f;
        const float csum = c3[o] + c5[o] + c7[o] + c11[o];
        for (int i = 0; i < 64; ++i) {
            const float y  = y0[b * 64 + i];
            const float Ti = (float)S_LEN * y + dt * sSC[i];

            #define TAPS(KPTR, KK, P)                                          \
                _Pragma("unroll")                                              \
                for (int k = 0; k < (KK); ++k) {                               \
                    const int dd = k - (P);                                    \
                    const float w = KPTR[(o * 64 + i) * (KK) + k];             \
                    float e;                                                   \
                    if (dd > 0)      e = dd  * y + dt * sHP[(dd - 1) * 64 + i];\
                    else if (dd < 0) e = -dd * y + dt * sTP[(-dd - 1) * 64 + i];\
                    else             e = 0.0f;                                 \
                    acc += w * (Ti - e);                                       \
                }
            TAPS(K3, 3, 1)
            TAPS(K5, 5, 2)
            TAPS(K7, 7, 3)
            TAPS(K11, 11, 5)
            #undef TAPS
        }
        sFeat[b * 64 + o] = 0.25f * (csum + acc * (1.0f / (float)S_LEN));
    }
    __syncthreads();

    if (tid < 64) {
        const int b = tid;
        float hacc = 0.0f;
        for (int j = 0; j < 64; ++j) {
            float s = hb1[j];
            for (int i = 0; i < 64; ++i) s += sFeat[b * 64 + i] * H1[i * 64 + j];
            s = fmaxf(s, 0.0f);
            hacc += s * H2[j];
        }
        hurst[b] = 1.0f / (1.0f + expf(-(hacc + hb2[0])));
    }
}

// ===========================================================================
extern "C" void kernel_launch(void* const* d_in, const int* in_sizes, int n_in,
                              void* d_out, int out_size, void* d_ws, size_t ws_size,
                              hipStream_t stream) {
    (void)in_sizes; (void)n_in; (void)out_size; (void)ws_size;

    const float* t   = (const float*)d_in[0];
    const float* y0  = (const float*)d_in[1];
    const float* W1  = (const float*)d_in[2];
    const float* b1  = (const float*)d_in[3];
    const float* W2  = (const float*)d_in[4];
    const float* b2  = (const float*)d_in[5];
    const float* W3  = (const float*)d_in[6];
    const float* b3  = (const float*)d_in[7];
    const float* W4  = (const float*)d_in[8];
    const float* b4  = (const float*)d_in[9];
    const float* K3  = (const float*)d_in[10];
    const float* c3  = (const float*)d_in[11];
    const float* K5  = (const float*)d_in[12];
    const float* c5  = (const float*)d_in[13];
    const float* K7  = (const float*)d_in[14];
    const float* c7  = (const float*)d_in[15];
    const float* K11 = (const float*)d_in[16];
    const float* c11 = (const float*)d_in[17];
    const float* H1  = (const float*)d_in[18];
    const float* hb1 = (const float*)d_in[19];
    const float* H2  = (const float*)d_in[20];
    const float* hb2 = (const float*)d_in[21];

    float* ws    = (float*)d_ws;
    float* RHS   = ws;                        // 8192*64
    float* Pbuf  = RHS  + S_LEN * D_DIM;      // 8192*64
    float* CSum  = Pbuf + S_LEN * D_DIM;      // 64*64
    float* CExcl = CSum + 64 * 64;            // 64*64
    float* Off   = CExcl + 64 * 64;           // 64*64
    float* SC    = Off  + 64 * 64;            // 64

    float* out   = (float*)d_out;
    float* hurst = out + (size_t)64 * S_LEN * D_DIM;

    mlp_rhs_kernel<<<256, 64, 0, stream>>>(t, W1, b1, W2, b2, W3, b3, W4, b4, RHS);
    scan_local_kernel<<<64, 64, 0, stream>>>(RHS, Pbuf, CSum, CExcl);
    scan_chunks_kernel<<<1, 64, 0, stream>>>(CSum, CExcl, Off, SC);
    write_sol_kernel<<<32768, 256, 0, stream>>>(t, y0, Pbuf, Off, out);
    features_kernel<<<1, 256, 0, stream>>>(t, y0, Pbuf, Off, SC,
                                           K3, c3, K5, c5, K7, c7, K11, c11,
                                           H1, hb1, H2, hb2, hurst);
}